// MambaEncoderBlock_81844896792906
// MI455X (gfx1250) — compile-verified
//
#include <hip/hip_runtime.h>
#include <hip/hip_bf16.h>

// ---------------------------------------------------------------------------
// Mamba encoder block for MI455X (gfx1250, wave32, WMMA f32 16x16x4)
// B=4, L=2048, D_MODEL=512, D_INNER=1024, D_STATE=16, D_CONV=4, DT_RANK=32
// ---------------------------------------------------------------------------

#define B_DIM 4
#define L_DIM 2048
#define D_MODEL 512
#define D_INNER 1024
#define D_STATE 16
#define D_CONV 4
#define DT_RANK 32
#define ROWS (B_DIM * L_DIM)          // 8192
#define N_XZ (2 * D_INNER)            // 2048
#define N_DBL (DT_RANK + 2 * D_STATE) // 64
#define EPS 1e-6f

// GEMM tiling: 256 threads = 8 waves; block tile 128(M) x 64(N); K-chunk 32.
#define KC 32
#define BN 64
#define LDS_STRIDE 34 // padded transposed B tile: sB[n*34 + k], conflict-free

typedef __attribute__((ext_vector_type(2))) float v2f;
typedef __attribute__((ext_vector_type(8))) float v8f;

__device__ __forceinline__ float silu_f(float v) {
    return v / (1.0f + __expf(-v));
}
__device__ __forceinline__ float softplus_f(float v) {
    return (v > 20.0f) ? v : log1pf(__expf(v));
}

// ---------------------------------------------------------------------------
// Kernel 1: per-row RMS scale  rs[row] = rsqrt(mean(x^2) + eps)
// ---------------------------------------------------------------------------
__global__ void rms_scale_kernel(const float* __restrict__ x,
                                 float* __restrict__ rs) {
    __shared__ float red[256];
    const int row = blockIdx.x;
    const float* xr = x + (size_t)row * D_MODEL;
    float s = 0.0f;
    for (int i = threadIdx.x; i < D_MODEL; i += 256) {
        float v = xr[i];
        s += v * v;
    }
    red[threadIdx.x] = s;
    __syncthreads();
    for (int off = 128; off > 0; off >>= 1) {
        if (threadIdx.x < off) red[threadIdx.x] += red[threadIdx.x + off];
        __syncthreads();
    }
    if (threadIdx.x == 0)
        rs[row] = rsqrtf(red[0] * (1.0f / D_MODEL) + EPS);
}

// ---------------------------------------------------------------------------
// Block-level WMMA GEMM core.
// 8 waves, each computing a 16(M) x 64(N) tile; the block covers 128 x 64.
// B chunks (KC x 64) are staged once per block into LDS, transposed with a
// 34-dword stride:  sB[n*34 + k].  Fragment reads are aligned ds_load_b64
// and bank-conflict-free; staging writes are conflict-free too.
// Fragment layouts per CDNA5 ISA 7.12.2 (f32 16x16x4):
//   A lane t: m = t&15, k = k0 + 2*(t>>4) + {0,1}
//   B lane t: n = t&15, k = k0 + 2*(t>>4) + {0,1}
//   D lane t, vgpr v: m = v + 8*(t>>4), n = t&15
// ---------------------------------------------------------------------------
template <bool SCALE_A>
__device__ __forceinline__ void gemm_block_core(
    const float* __restrict__ A, int lda,
    const float* __restrict__ Bm, int ldb, int K,
    const float* __restrict__ a_col_scale, // norm_w if SCALE_A
    const float* __restrict__ rs_vec,      // per-row scale if SCALE_A
    int m_block, int n0, float* __restrict__ sB, v8f acc[4]) {
    const int tid = threadIdx.x;
    const int lane = tid & 31;
    const int wave = tid >> 5;
    const int mlane = lane & 15;
    const int khalf = (lane >> 4) << 1; // 0 or 2
    const int m0 = m_block + wave * 16;
    const float* Arow = A + (size_t)(m0 + mlane) * lda;
    const float rscale = SCALE_A ? rs_vec[m0 + mlane] : 1.0f;

    for (int kc = 0; kc < K; kc += KC) {
        __syncthreads(); // previous chunk's LDS reads complete
#pragma unroll
        for (int i = 0; i < (KC * BN) / 256; ++i) { // 8 coalesced loads/thread
            const int idx = i * 256 + tid;
            const int kr = idx >> 6;  // 0..31
            const int nc = idx & 63;  // 0..63
            sB[nc * LDS_STRIDE + kr] =
                Bm[(size_t)(kc + kr) * ldb + (n0 + nc)];
        }
        __builtin_prefetch(Arow + kc + KC, 0, 1); // next A chunk into caches
        __syncthreads();
#pragma unroll
        for (int kk = 0; kk < KC; kk += 4) {
            const int ka = kc + kk + khalf;
            v2f a;
            a.x = Arow[ka];
            a.y = Arow[ka + 1];
            if (SCALE_A) {
                a.x *= a_col_scale[ka] * rscale;
                a.y *= a_col_scale[ka + 1] * rscale;
            }
#pragma unroll
            for (int j = 0; j < 4; ++j) {
                const int nl = j * 16 + mlane;
                const v2f b =
                    *(const v2f*)&sB[nl * LDS_STRIDE + kk + khalf];
                acc[j] = __builtin_amdgcn_wmma_f32_16x16x4_f32(
                    false, a, false, b, (short)0, acc[j], false, false);
            }
        }
    }
}

// ---------------------------------------------------------------------------
// Kernel 2: xz[8192,2048] = (x * rs[m] * norm_w[k]) @ W_in   (RMSNorm fused)
// ---------------------------------------------------------------------------
__global__ void gemm_in_kernel(const float* __restrict__ x,
                               const float* __restrict__ rs,
                               const float* __restrict__ norm_w,
                               const float* __restrict__ W_in,
                               float* __restrict__ xz) {
    __shared__ float sB[BN * LDS_STRIDE];
    const int lane = threadIdx.x & 31;
    const int wave = threadIdx.x >> 5;
    const int m_block = blockIdx.y * 128;
    const int n0 = blockIdx.x * 64;
    v8f acc[4] = {};
    gemm_block_core<true>(x, D_MODEL, W_in, N_XZ, D_MODEL, norm_w, rs,
                          m_block, n0, sB, acc);
    const int mlane = lane & 15;
    const int mbase = m_block + wave * 16 + ((lane >> 4) << 3);
#pragma unroll
    for (int j = 0; j < 4; ++j) {
        const int n = n0 + j * 16 + mlane;
#pragma unroll
        for (int v = 0; v < 8; ++v)
            xz[(size_t)(mbase + v) * N_XZ + n] = acc[j][v];
    }
}

// ---------------------------------------------------------------------------
// Kernel 3: causal depthwise conv1d over L + SiLU
//   xc[b,l,d] = silu(conv_b[d] + sum_k conv_w[d,k]*x_in[b,l-3+k,d])
//   x_in = xz[:, 0:1024] (row stride 2048)
// ---------------------------------------------------------------------------
__global__ void conv_silu_kernel(const float* __restrict__ xz,
                                 const float* __restrict__ conv_w,
                                 const float* __restrict__ conv_b,
                                 float* __restrict__ xc) {
    const int idx = blockIdx.x * 256 + threadIdx.x; // [0, ROWS*D_INNER)
    const int d = idx & (D_INNER - 1);
    const int row = idx >> 10; // b*L + l
    const int l = row & (L_DIM - 1);
    float acc = conv_b[d];
#pragma unroll
    for (int k = 0; k < D_CONV; ++k) {
        const int lk = l - (D_CONV - 1) + k;
        if (lk >= 0)
            acc += conv_w[d * D_CONV + k] *
                   xz[(size_t)(row - (D_CONV - 1) + k) * N_XZ + d];
    }
    xc[idx] = silu_f(acc);
}

// ---------------------------------------------------------------------------
// Kernel 4: x_dbl[8192,64] = xc @ W_xproj   (K=1024)
// ---------------------------------------------------------------------------
__global__ void gemm_xproj_kernel(const float* __restrict__ xc,
                                  const float* __restrict__ W_xproj,
                                  float* __restrict__ xdbl) {
    __shared__ float sB[BN * LDS_STRIDE];
    const int lane = threadIdx.x & 31;
    const int wave = threadIdx.x >> 5;
    const int m_block = blockIdx.y * 128;
    v8f acc[4] = {};
    gemm_block_core<false>(xc, D_INNER, W_xproj, N_DBL, D_INNER, nullptr,
                           nullptr, m_block, 0, sB, acc);
    const int mlane = lane & 15;
    const int mbase = m_block + wave * 16 + ((lane >> 4) << 3);
#pragma unroll
    for (int j = 0; j < 4; ++j) {
        const int n = j * 16 + mlane;
#pragma unroll
        for (int v = 0; v < 8; ++v)
            xdbl[(size_t)(mbase + v) * N_DBL + n] = acc[j][v];
    }
}

// ---------------------------------------------------------------------------
// Kernel 5: dt[8192,1024] = softplus(x_dbl[:,0:32] @ W_dt + b_dt)  (K=32)
// ---------------------------------------------------------------------------
__global__ void gemm_dt_kernel(const float* __restrict__ xdbl,
                               const float* __restrict__ W_dt,
                               const float* __restrict__ b_dt,
                               float* __restrict__ dt) {
    __shared__ float sB[BN * LDS_STRIDE];
    const int lane = threadIdx.x & 31;
    const int wave = threadIdx.x >> 5;
    const int m_block = blockIdx.y * 128;
    const int n0 = blockIdx.x * 64;
    v8f acc[4] = {};
    gemm_block_core<false>(xdbl, N_DBL, W_dt, D_INNER, DT_RANK, nullptr,
                           nullptr, m_block, n0, sB, acc);
    const int mlane = lane & 15;
    const int mbase = m_block + wave * 16 + ((lane >> 4) << 3);
#pragma unroll
    for (int j = 0; j < 4; ++j) {
        const int n = n0 + j * 16 + mlane;
        const float bias = b_dt[n];
#pragma unroll
        for (int v = 0; v < 8; ++v)
            dt[(size_t)(mbase + v) * D_INNER + n] =
                softplus_f(acc[j][v] + bias);
    }
}

// ---------------------------------------------------------------------------
// Kernel 6: selective scan over L (serial critical path).
// grid (D_INNER/256, B), block 256; thread owns channel (b,d), h[16] in VGPRs.
// B_ssm/C_ssm staged through LDS for 8 timesteps per barrier pair (exactly
// one element per thread), so the 8 steps' dt/xc/z loads can be issued ahead
// of the recurrence.  Fuses +D_skip*xc and *silu(z).
// NOTE: y overwrites dt in place (each dt element is read exactly once by its
// owning thread before the overwrite in the same iteration).
// ---------------------------------------------------------------------------
#define SCAN_CH 8
__global__ void scan_kernel(const float* __restrict__ xz,
                            const float* __restrict__ xc,
                            const float* __restrict__ xdbl,
                            float* __restrict__ dt_y, // dt in, y out (aliased)
                            const float* __restrict__ A_log,
                            const float* __restrict__ D_skip) {
    __shared__ float sBC[SCAN_CH][2 * D_STATE]; // [t][0:16)=B, [16:32)=C
    const int b = blockIdx.y;
    const int d = blockIdx.x * 256 + threadIdx.x;
    float Arow[D_STATE];
#pragma unroll
    for (int s = 0; s < D_STATE; ++s)
        Arow[s] = -__expf(A_log[d * D_STATE + s]);
    const float Dv = D_skip[d];
    float h[D_STATE];
#pragma unroll
    for (int s = 0; s < D_STATE; ++s) h[s] = 0.0f;

    for (int lc = 0; lc < L_DIM; lc += SCAN_CH) {
        __syncthreads();
        { // 256 threads load SCAN_CH*32 = 256 floats: one each, no divergence
            const int t = threadIdx.x >> 5; // timestep within chunk
            const int s = threadIdx.x & 31; // element within B||C
            sBC[t][s] =
                xdbl[(size_t)(b * L_DIM + lc + t) * N_DBL + DT_RANK + s];
        }
        __syncthreads();
#pragma unroll
        for (int u = 0; u < SCAN_CH; ++u) {
            const int row = b * L_DIM + lc + u;
            const size_t cidx = (size_t)row * D_INNER + d;
            const float dt = dt_y[cidx];
            const float xcv = xc[cidx];
            const float z = xz[(size_t)row * N_XZ + D_INNER + d];
            float y = 0.0f;
#pragma unroll
            for (int s = 0; s < D_STATE; ++s) {
                const float dA = __expf(dt * Arow[s]);
                h[s] = dA * h[s] + (dt * sBC[u][s]) * xcv;
                y += h[s] * sBC[u][D_STATE + s];
            }
            y += Dv * xcv;
            y *= silu_f(z);
            dt_y[cidx] = y;
        }
    }
}

// ---------------------------------------------------------------------------
// Kernel 7: out[8192,512] = y @ W_out + residual(x)   (K=1024)
// ---------------------------------------------------------------------------
__global__ void gemm_out_kernel(const float* __restrict__ y,
                                const float* __restrict__ W_out,
                                const float* __restrict__ x,
                                float* __restrict__ out) {
    __shared__ float sB[BN * LDS_STRIDE];
    const int lane = threadIdx.x & 31;
    const int wave = threadIdx.x >> 5;
    const int m_block = blockIdx.y * 128;
    const int n0 = blockIdx.x * 64;
    v8f acc[4] = {};
    gemm_block_core<false>(y, D_INNER, W_out, D_MODEL, D_INNER, nullptr,
                           nullptr, m_block, n0, sB, acc);
    const int mlane = lane & 15;
    const int mbase = m_block + wave * 16 + ((lane >> 4) << 3);
#pragma unroll
    for (int j = 0; j < 4; ++j) {
        const int n = n0 + j * 16 + mlane;
#pragma unroll
        for (int v = 0; v < 8; ++v) {
            const size_t o = (size_t)(mbase + v) * D_MODEL + n;
            out[o] = acc[j][v] + x[o];
        }
    }
}

// ---------------------------------------------------------------------------
// Launch
// ---------------------------------------------------------------------------
extern "C" void kernel_launch(void* const* d_in, const int* in_sizes, int n_in,
                              void* d_out, int out_size, void* d_ws,
                              size_t ws_size, hipStream_t stream) {
    const float* x      = (const float*)d_in[0];
    const float* norm_w = (const float*)d_in[1];
    const float* W_in   = (const float*)d_in[2];
    const float* conv_w = (const float*)d_in[3];
    const float* conv_b = (const float*)d_in[4];
    const float* W_xprj = (const float*)d_in[5];
    const float* W_dt   = (const float*)d_in[6];
    const float* b_dt   = (const float*)d_in[7];
    const float* A_log  = (const float*)d_in[8];
    const float* D_skip = (const float*)d_in[9];
    const float* W_out  = (const float*)d_in[10];
    float* out = (float*)d_out;

    // Workspace layout (floats). dt buffer doubles as y buffer (see scan).
    float* ws = (float*)d_ws;
    float* rs   = ws;                           // 8192
    float* xz   = rs + ROWS;                    // 8192*2048
    float* xc   = xz + (size_t)ROWS * N_XZ;     // 8192*1024
    float* xdbl = xc + (size_t)ROWS * D_INNER;  // 8192*64
    float* dt_y = xdbl + (size_t)ROWS * N_DBL;  // 8192*1024

    // 1) RMS row scales
    rms_scale_kernel<<<ROWS, 256, 0, stream>>>(x, rs);

    // 2) xz = norm(x) @ W_in
    gemm_in_kernel<<<dim3(N_XZ / 64, ROWS / 128), 256, 0, stream>>>(
        x, rs, norm_w, W_in, xz);

    // 3) depthwise causal conv + SiLU
    conv_silu_kernel<<<(ROWS * D_INNER) / 256, 256, 0, stream>>>(
        xz, conv_w, conv_b, xc);

    // 4) x_dbl = xc @ W_xproj
    gemm_xproj_kernel<<<dim3(1, ROWS / 128), 256, 0, stream>>>(xc, W_xprj,
                                                               xdbl);

    // 5) dt = softplus(x_dbl[:, :32] @ W_dt + b_dt)
    gemm_dt_kernel<<<dim3(D_INNER / 64, ROWS / 128), 256, 0, stream>>>(
        xdbl, W_dt, b_dt, dt_y);

    // 6) selective scan (+D_skip*xc, *silu(z)); overwrites dt with y
    scan_kernel<<<dim3(D_INNER / 256, B_DIM), 256, 0, stream>>>(
        xz, xc, xdbl, dt_y, A_log, D_skip);

    // 7) out = y @ W_out + residual
    gemm_out_kernel<<<dim3(D_MODEL / 64, ROWS / 128), 256, 0, stream>>>(
        dt_y, W_out, x, out);
}